// grouped_query_attn_5643587027492
// MI455X (gfx1250) — compile-verified
//
#include <hip/hip_runtime.h>
#include <hip/hip_bf16.h>
#include <math.h>

typedef __attribute__((ext_vector_type(2))) float v2f;
typedef __attribute__((ext_vector_type(8))) float v8f;
typedef __attribute__((ext_vector_type(4))) int   v4i;

#define N_EMBD 2048
#define NH 16
#define NKV 4
#define DH 128
#define BB 4
#define TT 2048
#define NP 1152                 // packed N: 128 (q-sum) + 512 (k) + 512 (v)
#define KVW (NKV * DH)          // 512
#define SCALE 0.08838834764831845f  // 1/sqrt(128)

// gfx1250 async global->LDS path (ASYNCcnt), guarded so the fallback still
// compiles if this toolchain does not declare the builtins.
#if defined(__has_builtin)
#if __has_builtin(__builtin_amdgcn_global_load_async_to_lds_b128) && \
    __has_builtin(__builtin_amdgcn_s_wait_asynccnt)
#define HAVE_ASYNC_LDS 1
#endif
#endif
#ifndef HAVE_ASYNC_LDS
#define HAVE_ASYNC_LDS 0
#endif

static __device__ __forceinline__ v8f wmma4(v2f a, v2f b, v8f c) {
  // V_WMMA_F32_16X16X4_F32: full fp32 matrix op (D = A[16x4] * B[4x16] + C)
  return __builtin_amdgcn_wmma_f32_16x16x4_f32(
      /*neg_a=*/false, a, /*neg_b=*/false, b,
      /*c_mod=*/(short)0, c, /*reuse_a=*/false, /*reuse_b=*/false);
}

#if HAVE_ASYNC_LDS
typedef v4i __attribute__((address_space(1)))* gptr4;
typedef v4i __attribute__((address_space(3)))* lptr4;
// Copy 64 contiguous bytes global -> LDS per thread via 4 async b128 ops.
// INST_OFFSET is added to both the global and LDS addresses (ISA ch.10 async).
static __device__ __forceinline__ void stage_async64(const float* gsrc, float* ldst) {
  gptr4 g = (gptr4)(void*)const_cast<float*>(gsrc);
  lptr4 l = (lptr4)(void*)ldst;
  __builtin_amdgcn_global_load_async_to_lds_b128(g, l, 0, 0);
  __builtin_amdgcn_global_load_async_to_lds_b128(g, l, 16, 0);
  __builtin_amdgcn_global_load_async_to_lds_b128(g, l, 32, 0);
  __builtin_amdgcn_global_load_async_to_lds_b128(g, l, 48, 0);
}
#endif

// ---------------------------------------------------------------------------
// Pack W: rows 0..127 = sum over 16 q-heads (scaled by 1/sqrt(DH)),
//         rows 128..1151 = k,v rows of W (W rows 2048..3071 => +1920).
// ---------------------------------------------------------------------------
__global__ void pack_w(const float* __restrict__ W, float* __restrict__ Wp) {
  int idx = blockIdx.x * 256 + threadIdx.x;   // over NP * N_EMBD
  int r = idx >> 11;
  int c = idx & (N_EMBD - 1);
  float v;
  if (r < DH) {
    v = 0.0f;
    #pragma unroll
    for (int h = 0; h < NH; ++h) v += W[(size_t)(h * DH + r) * N_EMBD + c];
    v *= SCALE;
  } else {
    v = W[(size_t)(r + 1920) * N_EMBD + c];
  }
  Wp[(size_t)r * N_EMBD + c] = v;
}

__global__ void pack_b(const float* __restrict__ b, float* __restrict__ bp) {
  int r = blockIdx.x * 256 + threadIdx.x;
  if (r >= NP) return;
  float v;
  if (r < DH) {
    v = 0.0f;
    #pragma unroll
    for (int h = 0; h < NH; ++h) v += b[h * DH + r];
    v *= SCALE;
  } else {
    v = b[r + 1920];
  }
  bp[r] = v;
}

// ---------------------------------------------------------------------------
// QKV GEMM:  C[8192 x 1152] = X[8192 x 2048] * Wp^T + bp
// Block: 256 thr = 8 waves (4 M-waves x 2 N-waves); macro-tile 128M x 64N.
// Each wave: 32x32 => 2x2 f32-WMMA accumulators. LDS staged, pad-36 stride
// (conflict-free + 8B-aligned v2f fragment loads -> ds_load_2addr_b64).
// ---------------------------------------------------------------------------
#define KT  32
#define LDA 36
__global__ __launch_bounds__(256) void qkv_gemm(
    const float* __restrict__ X, const float* __restrict__ Wp,
    const float* __restrict__ bp, float* __restrict__ qs,
    float* __restrict__ kbuf, float* __restrict__ vbuf) {
  __shared__ float Asm[128][LDA];
  __shared__ float Bsm[64][LDA];

  const int tid  = threadIdx.x;
  const int w    = tid >> 5;
  const int lane = tid & 31;
  const int lm   = lane & 15;
  const int hi   = lane >> 4;
  const int wm   = (w & 3) * 32;
  const int wn   = (w >> 2) * 32;
  const int M0   = blockIdx.y * 128;
  const int N0   = blockIdx.x * 64;

  v8f acc[2][2];
  acc[0][0] = (v8f){}; acc[0][1] = (v8f){};
  acc[1][0] = (v8f){}; acc[1][1] = (v8f){};

  const int arow  = tid >> 1;          // 0..127
  const int ahalf = (tid & 1) * 16;    // 0 or 16
  const int brow  = tid >> 2;          // 0..63
  const int bq    = (tid & 3) * 8;     // 0,8,16,24

  for (int k0 = 0; k0 < N_EMBD; k0 += KT) {
    const float* xsrc = X + (size_t)(M0 + arow) * N_EMBD + k0 + ahalf;
    float4 xa0 = *(const float4*)(xsrc + 0);
    float4 xa1 = *(const float4*)(xsrc + 4);
    float4 xa2 = *(const float4*)(xsrc + 8);
    float4 xa3 = *(const float4*)(xsrc + 12);
    const float* wsrc = Wp + (size_t)(N0 + brow) * N_EMBD + k0 + bq;
    float4 wb0 = *(const float4*)(wsrc + 0);
    float4 wb1 = *(const float4*)(wsrc + 4);

    __syncthreads();
    *(float4*)&Asm[arow][ahalf + 0]  = xa0;
    *(float4*)&Asm[arow][ahalf + 4]  = xa1;
    *(float4*)&Asm[arow][ahalf + 8]  = xa2;
    *(float4*)&Asm[arow][ahalf + 12] = xa3;
    *(float4*)&Bsm[brow][bq + 0]     = wb0;
    *(float4*)&Bsm[brow][bq + 4]     = wb1;
    __syncthreads();

    #pragma unroll
    for (int kk = 0; kk < KT; kk += 4) {
      // A frag (16x4 f32): lane = M, vgpr0/1 = K {2*hi, 2*hi+1}
      v2f a0 = *(const v2f*)&Asm[wm + lm][kk + 2 * hi];
      v2f a1 = *(const v2f*)&Asm[wm + 16 + lm][kk + 2 * hi];
      // B frag (4x16 f32): lane = N, vgpr0/1 = K {2*hi, 2*hi+1}; B[k][n] = Wp[n][k]
      v2f b0 = *(const v2f*)&Bsm[wn + lm][kk + 2 * hi];
      v2f b1 = *(const v2f*)&Bsm[wn + 16 + lm][kk + 2 * hi];
      acc[0][0] = wmma4(a0, b0, acc[0][0]);
      acc[0][1] = wmma4(a0, b1, acc[0][1]);
      acc[1][0] = wmma4(a1, b0, acc[1][0]);
      acc[1][1] = wmma4(a1, b1, acc[1][1]);
    }
  }

  // Epilogue: bias + scatter into qs / k / v regions.
  #pragma unroll
  for (int sm = 0; sm < 2; ++sm) {
    #pragma unroll
    for (int sn = 0; sn < 2; ++sn) {
      int n = N0 + wn + sn * 16 + lm;
      float bias = bp[n];
      #pragma unroll
      for (int r = 0; r < 8; ++r) {
        int m = M0 + wm + sm * 16 + r + 8 * hi;   // C layout: vgpr r -> rows r / r+8
        float val = acc[sm][sn][r] + bias;
        if (n < DH)            qs[(size_t)m * DH + n] = val;
        else if (n < DH + KVW) kbuf[(size_t)m * KVW + (n - DH)] = val;
        else                   vbuf[(size_t)m * KVW + (n - DH - KVW)] = val;
      }
    }
  }
}

// ---------------------------------------------------------------------------
// Flash attention per (b, kv-head, 128-query block). 8 waves x 16 queries.
// Q frags resident in regs; K/V tiles (16 keys) double-buffered in LDS via
// async global->LDS copies (ASYNCcnt) when available; online softmax with
// width-16 cross-lane reductions (matches WMMA C-layout halves); P goes
// through per-wave LDS to convert C-layout -> A-layout for the PV WMMA.
// ---------------------------------------------------------------------------
#define LDK 132
#define NTILES (TT / 16)
__global__ __launch_bounds__(256) void gqa_attn(
    const float* __restrict__ qs, const float* __restrict__ kbuf,
    const float* __restrict__ vbuf, float* __restrict__ y) {
  __shared__ float Ksm[2][16][LDK];
  __shared__ float Vsm[2][16][LDK];
  __shared__ float Psm[8][16][18];

  const int tid  = threadIdx.x;
  const int w    = tid >> 5;
  const int lane = tid & 31;
  const int lm   = lane & 15;
  const int hi   = lane >> 4;

  const int qblocks = TT / 128;
  const int bh  = blockIdx.x / qblocks;
  const int qb  = blockIdx.x % qblocks;
  const int b   = bh / NKV;
  const int kvh = bh % NKV;
  const int qbase = qb * 128;
  const int qrow  = qbase + w * 16 + lm;

  // Load Q fragments once (1/sqrt(DH) scale already folded into qs).
  v2f qa[32];
  #pragma unroll
  for (int ks = 0; ks < 32; ++ks)
    qa[ks] = *(const v2f*)&qs[(size_t)(b * TT + qrow) * DH + ks * 4 + 2 * hi];

  v8f o[8];
  #pragma unroll
  for (int j = 0; j < 8; ++j) o[j] = (v8f){};
  float mrow[8], lrow[8];
  #pragma unroll
  for (int r = 0; r < 8; ++r) { mrow[r] = -INFINITY; lrow[r] = 0.0f; }

  const float* kb = kbuf + (size_t)b * TT * KVW + kvh * DH;
  const float* vb = vbuf + (size_t)b * TT * KVW + kvh * DH;

  const int srow  = (tid & 127) >> 3;  // 0..15
  const int sseg  = (tid & 7) * 16;    // 0..112
  const bool isV  = tid >= 128;
  const float* src = isV ? vb : kb;

#if HAVE_ASYNC_LDS
  // Prologue: async-stage tile 0 into buffer 0.
  {
    const float* p0 = src + (size_t)srow * KVW + sseg;
    float* d0 = isV ? &Vsm[0][srow][sseg] : &Ksm[0][srow][sseg];
    stage_async64(p0, d0);
  }
#endif

  for (int kt = 0; kt < NTILES; ++kt) {
    const int cur = kt & 1;

#if HAVE_ASYNC_LDS
    // Issue next tile into the other buffer (safe: end-of-iteration barrier
    // of kt-1 guarantees everyone finished computing on it), then wait for
    // the current tile (async loads complete in order => <=4 outstanding).
    if (kt + 1 < NTILES) {
      const float* pn = src + (size_t)((kt + 1) * 16 + srow) * KVW + sseg;
      float* dn = isV ? &Vsm[cur ^ 1][srow][sseg] : &Ksm[cur ^ 1][srow][sseg];
      stage_async64(pn, dn);
      __builtin_amdgcn_s_wait_asynccnt(4);
    } else {
      __builtin_amdgcn_s_wait_asynccnt(0);
    }
    __syncthreads();
#else
    {
      const float* p = src + (size_t)(kt * 16 + srow) * KVW + sseg;
      float4 t0 = *(const float4*)(p + 0);
      float4 t1 = *(const float4*)(p + 4);
      float4 t2 = *(const float4*)(p + 8);
      float4 t3 = *(const float4*)(p + 12);
      __syncthreads();
      float* dst = isV ? &Vsm[cur][srow][sseg] : &Ksm[cur][srow][sseg];
      *(float4*)(dst + 0)  = t0;
      *(float4*)(dst + 4)  = t1;
      *(float4*)(dst + 8)  = t2;
      *(float4*)(dst + 12) = t3;
      __syncthreads();
    }
#endif

    // S[16q x 16k] = Q * K^T : B[k][n] = Ksm[n][k]
    v8f s = (v8f){};
    #pragma unroll
    for (int ks = 0; ks < 32; ++ks) {
      v2f bf = *(const v2f*)&Ksm[cur][lm][ks * 4 + 2 * hi];
      s = wmma4(qa[ks], bf, s);
    }

    // Online softmax update per row slot (rows r and r+8 live in lane halves).
    #pragma unroll
    for (int r = 0; r < 8; ++r) {
      float sv = s[r];
      float rm = sv;
      rm = fmaxf(rm, __shfl_xor(rm, 1, 16));
      rm = fmaxf(rm, __shfl_xor(rm, 2, 16));
      rm = fmaxf(rm, __shfl_xor(rm, 4, 16));
      rm = fmaxf(rm, __shfl_xor(rm, 8, 16));
      float mnew = fmaxf(mrow[r], rm);
      float f    = __expf(mrow[r] - mnew);
      float pv   = __expf(sv - mnew);
      float rs = pv;
      rs += __shfl_xor(rs, 1, 16);
      rs += __shfl_xor(rs, 2, 16);
      rs += __shfl_xor(rs, 4, 16);
      rs += __shfl_xor(rs, 8, 16);
      lrow[r] = lrow[r] * f + rs;
      mrow[r] = mnew;
      #pragma unroll
      for (int j = 0; j < 8; ++j) o[j][r] *= f;
      Psm[w][r + 8 * hi][lm] = pv;   // P stored [query][key]
    }
    __syncthreads();   // make P visible before A-layout reads

    // O[16q x 128d] += P[16x16] * V[16x128]
    #pragma unroll
    for (int ks = 0; ks < 4; ++ks) {
      v2f a = *(const v2f*)&Psm[w][lm][ks * 4 + 2 * hi];
      #pragma unroll
      for (int j = 0; j < 8; ++j) {
        v2f bf;
        bf.x = Vsm[cur][ks * 4 + 2 * hi][j * 16 + lm];
        bf.y = Vsm[cur][ks * 4 + 2 * hi + 1][j * 16 + lm];
        o[j] = wmma4(a, bf, o[j]);
      }
    }

#if HAVE_ASYNC_LDS
    __syncthreads();   // release buffer `cur` before it is rewritten at kt+2
#endif
  }

  // Normalize and store: y[b, t, kvh*128 + d]
  #pragma unroll
  for (int r = 0; r < 8; ++r) {
    float inv = 1.0f / lrow[r];
    int t = qbase + w * 16 + r + 8 * hi;
    float* dst = y + (size_t)(b * TT + t) * KVW + kvh * DH;
    #pragma unroll
    for (int j = 0; j < 8; ++j) dst[j * 16 + lm] = o[j][r] * inv;
  }
}

// ---------------------------------------------------------------------------
extern "C" void kernel_launch(void* const* d_in, const int* in_sizes, int n_in,
                              void* d_out, int out_size, void* d_ws, size_t ws_size,
                              hipStream_t stream) {
  const float* x    = (const float*)d_in[0];
  const float* W    = (const float*)d_in[1];
  const float* bias = (const float*)d_in[2];
  float* out = (float*)d_out;

  float* ws = (float*)d_ws;
  float* Wp = ws;                                   // 1152*2048
  float* bp = Wp + (size_t)NP * N_EMBD;             // 1152 (pad to 2048)
  float* qs = bp + 2048;                            // 8192*128
  float* kb = qs + (size_t)BB * TT * DH;            // 8192*512
  float* vb = kb + (size_t)BB * TT * KVW;           // 8192*512

  pack_w<<<(NP * N_EMBD) / 256, 256, 0, stream>>>(W, Wp);
  pack_b<<<(NP + 255) / 256, 256, 0, stream>>>(bias, bp);
  qkv_gemm<<<dim3(NP / 64, (BB * TT) / 128), 256, 0, stream>>>(x, Wp, bp, qs, kb, vb);
  gqa_attn<<<BB * NKV * (TT / 128), 256, 0, stream>>>(qs, kb, vb, out);
}